// Sampler_58849641890404
// MI455X (gfx1250) — compile-verified
//
#include <hip/hip_runtime.h>
#include <hip/hip_bf16.h>
#include <stdint.h>

#define BROWS 128
#define VCOLS 128000
#define CHUNK 6400          // floats per TDM tile (25.6 KB); 2 tiles ping-pong
#define NCHUNK 20           // 20 * 6400 = 128000
#define NTHREADS 512        // 16 wave32s per row

typedef unsigned int u32;
typedef u32 u32x4 __attribute__((ext_vector_type(4)));
typedef int  i32x4 __attribute__((ext_vector_type(4)));
typedef int  i32x8 __attribute__((ext_vector_type(8)));

// ---- monotonic float -> uint key (ascending key == ascending float) ----
__device__ __forceinline__ u32 monokey(float f) {
    u32 u = __float_as_uint(f);
    return (u & 0x80000000u) ? ~u : (u | 0x80000000u);
}

__device__ __forceinline__ u32 rotl32(u32 x, int r) { return (x << r) | (x >> (32 - r)); }

// ---- exact Threefry-2x32 matching jax.random.exponential(jax.random.key(1), (B,V)) ----
__device__ __forceinline__ u32 threefry_bits(u32 i) {
    const u32 NH = (u32)BROWS * (u32)VCOLS / 2u;  // 8,192,000
    u32 c0, c1; bool hi;
    if (i < NH) { c0 = i; c1 = i + NH; hi = false; }
    else        { c0 = i - NH; c1 = i; hi = true; }
    const u32 ks0 = 0u, ks1 = 1u, ks2 = 0x1BD11BDAu ^ 0u ^ 1u;
    u32 x0 = c0 + ks0, x1 = c1 + ks1;
#define TFR(r) { x0 += x1; x1 = rotl32(x1, (r)); x1 ^= x0; }
    TFR(13) TFR(15) TFR(26) TFR(6)   x0 += ks1; x1 += ks2 + 1u;
    TFR(17) TFR(29) TFR(16) TFR(24)  x0 += ks2; x1 += ks0 + 2u;
    TFR(13) TFR(15) TFR(26) TFR(6)   x0 += ks0; x1 += ks1 + 3u;
    TFR(17) TFR(29) TFR(16) TFR(24)  x0 += ks1; x1 += ks2 + 4u;
    TFR(13) TFR(15) TFR(26) TFR(6)   x0 += ks2; x1 += ks0 + 5u;
#undef TFR
    return hi ? x1 : x0;
}

__device__ __forceinline__ float expo_of(u32 i) {
    u32 b = threefry_bits(i);
    float u = __uint_as_float((b >> 9) | 0x3F800000u) - 1.0f;  // [0,1)
    float e = -log1pf(-u);
    return fmaxf(e, 1e-10f);
}

// ---- TDM: DMA one 1-D tile of `nelem` floats from global into LDS ----
__device__ __forceinline__ void tdm_load_chunk(const float* gsrc, u32 lds_byte_addr, int nelem) {
    unsigned long long ga = (unsigned long long)(uintptr_t)gsrc;
    u32x4 g0;
    g0.x = 1u;                                            // count=1 (valid), no gather
    g0.y = lds_byte_addr;                                 // lds_addr (bytes)
    g0.z = (u32)(ga & 0xFFFFFFFFull);                     // global_addr[31:0]
    g0.w = (u32)((ga >> 32) & 0x1FFFFFFull) | (2u << 30); // global_addr[56:32] | type=2
    i32x8 g1;
    u32 td0 = (u32)nelem, td1 = 1u, tl0 = (u32)nelem;
    g1[0] = (int)(2u << 16);                              // wg_mask=0, data_size=2 (4B)
    g1[1] = (int)((td0 & 0xFFFFu) << 16);
    g1[2] = (int)((td0 >> 16) | ((td1 & 0xFFFFu) << 16));
    g1[3] = (int)((td1 >> 16) | (tl0 << 16));
    g1[4] = 1;                                            // tile_dim1=1, tile_dim2=0
    g1[5] = (int)td0;                                     // tensor_dim0_stride
    g1[6] = 0; g1[7] = 0;
    i32x4 z4 = {0, 0, 0, 0};
#if defined(__clang_major__) && (__clang_major__ >= 23)
    i32x8 z8 = {0, 0, 0, 0, 0, 0, 0, 0};
    __builtin_amdgcn_tensor_load_to_lds(g0, g1, z4, z4, z8, 0);
#else
    __builtin_amdgcn_tensor_load_to_lds(g0, g1, z4, z4, 0);
#endif
}

__global__ __launch_bounds__(NTHREADS)
void sampler_kernel(const float* __restrict__ logits,
                    const float* __restrict__ temps,
                    const int*   __restrict__ top_ks,
                    const float* __restrict__ top_ps,
                    float*       __restrict__ out)
{
    __shared__ float s_tile[2][CHUNK];   // ping-pong TDM tiles
    __shared__ u32   s_hcnt[256];
    __shared__ float s_hmass[256];
    __shared__ u32   s_scanu[257];       // suffix-inclusive count scan (+sentinel)
    __shared__ float s_scanf[257];       // suffix-inclusive mass scan (+sentinel)
    __shared__ float s_redf[NTHREADS];
    __shared__ int   s_redi[NTHREADS];
    __shared__ float s_M, s_arem, s_aremn;
    __shared__ u32   s_prefix, s_krem, s_tk, s_tcut;
    __shared__ int   s_seli, s_selfb;

    const int row = blockIdx.x;
    const int tid = threadIdx.x;
    const float* rowp = logits + (size_t)row * VCOLS;
    const float4* p4  = (const float4*)rowp;
    const float invT  = 1.0f / temps[row];

    // =========== Pass 0 (HBM via double-buffered TDM): row max + L0 count hist ===========
    float lmax = -__builtin_inff();
    if (tid < 256) s_hcnt[tid] = 0u;
    if (tid < 32) tdm_load_chunk(rowp, (u32)(uintptr_t)&s_tile[0][0], CHUNK);  // prologue
    __syncthreads();
    for (int c = 0; c < NCHUNK; ++c) {
        if (tid < 32) {   // wave 0 drives the tensor DMA pipeline
            if (c + 1 < NCHUNK) {
                tdm_load_chunk(rowp + (c + 1) * CHUNK,
                               (u32)(uintptr_t)&s_tile[(c + 1) & 1][0], CHUNK);
                __builtin_amdgcn_s_wait_tensorcnt(1);   // oldest (chunk c) complete
            } else {
                __builtin_amdgcn_s_wait_tensorcnt(0);
            }
        }
        __syncthreads();
        const float* tp = &s_tile[c & 1][0];
        for (int i = tid; i < CHUNK; i += NTHREADS) {
            float x = tp[i] * invT;
            lmax = fmaxf(lmax, x);
            atomicAdd(&s_hcnt[monokey(x) >> 24], 1u);
        }
        __syncthreads();   // buffer (c&1) free for chunk c+2's DMA
    }
    // parallel max reduction
    s_redf[tid] = lmax;
    __syncthreads();
    for (int off = NTHREADS / 2; off > 0; off >>= 1) {
        if (tid < off) { float o = s_redf[tid + off]; if (o > s_redf[tid]) s_redf[tid] = o; }
        __syncthreads();
    }
    if (tid == 0) {
        s_M = s_redf[0];
        int kk = top_ks[row]; if (kk < 1) kk = 1; if (kk > VCOLS) kk = VCOLS;
        s_krem   = (u32)kk;
        s_prefix = 0u;
    }
    __syncthreads();

    // =========== Top-k radix descent: 4 levels (level 0 hist already built) ===========
    for (int lvl = 0; lvl <= 3; ++lvl) {
        const int shiftHi = 32 - 8 * lvl;   // lvl 0: compare is trivially true
        if (lvl > 0) {
            if (tid < 256) s_hcnt[tid] = 0u;
            __syncthreads();
            const u32 pref = s_prefix;
            for (int j = tid; j < VCOLS / 4; j += NTHREADS) {
                float4 v = p4[j];
                float xs[4] = {v.x, v.y, v.z, v.w};
#pragma unroll
                for (int c = 0; c < 4; ++c) {
                    u32 key = monokey(xs[c] * invT);
                    if ((key >> shiftHi) == pref)
                        atomicAdd(&s_hcnt[(key >> (shiftHi - 8)) & 0xFFu], 1u);
                }
            }
            __syncthreads();
        }
        // parallel suffix-inclusive scan of counts
        if (tid < 256) s_scanu[tid] = s_hcnt[tid];
        if (tid == 0)  s_scanu[256] = 0u;
        __syncthreads();
        for (int off = 1; off < 256; off <<= 1) {
            u32 v = 0u;
            if (tid < 256 && tid + off < 256) v = s_scanu[tid + off];
            __syncthreads();
            if (tid < 256) s_scanu[tid] += v;
            __syncthreads();
        }
        // crossing selection: E < kr <= S  (exact for integers; unique writer)
        u32 kr = s_krem;
        __syncthreads();
        if (tid < 256) {
            u32 S = s_scanu[tid], E = s_scanu[tid + 1];
            if (S >= kr && E < kr) {
                s_prefix = (s_prefix << 8) | (u32)tid;
                s_krem   = kr - E;
            }
        }
        __syncthreads();
    }
    if (tid == 0) s_tk = s_prefix;   // exact key of k-th largest value
    __syncthreads();
    const u32 tk  = s_tk;
    const float M = s_M;

    // =========== Z (kept mass) + top-p level-0 exp-mass histogram ===========
    float lz = 0.0f;
    if (tid < 256) s_hmass[tid] = 0.0f;
    __syncthreads();
    for (int j = tid; j < VCOLS / 4; j += NTHREADS) {
        float4 v = p4[j];
        float xs[4] = {v.x, v.y, v.z, v.w};
#pragma unroll
        for (int c = 0; c < 4; ++c) {
            float x = xs[c] * invT;
            u32 key = monokey(x);
            if (key >= tk) {
                float m = __expf(x - M);
                lz += m;
                atomicAdd(&s_hmass[key >> 24], m);
            }
        }
    }
    // parallel Z reduction
    s_redf[tid] = lz;
    __syncthreads();
    for (int off = NTHREADS / 2; off > 0; off >>= 1) {
        if (tid < off) s_redf[tid] += s_redf[tid + off];
        __syncthreads();
    }
    if (tid == 0) {
        s_arem   = top_ps[row] * s_redf[0];   // pZ threshold
        s_prefix = 0u;
    }
    __syncthreads();

    // =========== Top-p radix descent: 4 levels (level 0 hist already built) ===========
    for (int lvl = 0; lvl <= 3; ++lvl) {
        const int shiftHi = 32 - 8 * lvl;
        if (lvl > 0) {
            if (tid < 256) s_hmass[tid] = 0.0f;
            __syncthreads();
            const u32 pref = s_prefix;
            for (int j = tid; j < VCOLS / 4; j += NTHREADS) {
                float4 v = p4[j];
                float xs[4] = {v.x, v.y, v.z, v.w};
#pragma unroll
                for (int c = 0; c < 4; ++c) {
                    float x = xs[c] * invT;
                    u32 key = monokey(x);
                    if (key >= tk && (key >> shiftHi) == pref)
                        atomicAdd(&s_hmass[(key >> (shiftHi - 8)) & 0xFFu], __expf(x - M));
                }
            }
            __syncthreads();
        }
        // parallel suffix-inclusive scan of masses
        if (tid < 256) s_scanf[tid] = s_hmass[tid];
        if (tid == 0)  { s_scanf[256] = 0.0f; s_seli = -1; s_selfb = 256; }
        __syncthreads();
        for (int off = 1; off < 256; off <<= 1) {
            float v = 0.0f;
            if (tid < 256 && tid + off < 256) v = s_scanf[tid + off];
            __syncthreads();
            if (tid < 256) s_scanf[tid] += v;
            __syncthreads();
        }
        float thr = s_arem;
        __syncthreads();
        if (tid < 256) {
            float S = s_scanf[tid], E = s_scanf[tid + 1];
            if (S >= thr && E < thr) { s_seli = tid; s_aremn = thr - E; }
        }
        __syncthreads();
        if (s_seli < 0 && tid < 256 && s_hmass[tid] > 0.0f)
            atomicMin(&s_selfb, tid);   // rounding fallback: lowest non-empty bin
        __syncthreads();
        if (tid == 0) {
            int sel; float an;
            if (s_seli >= 0) { sel = s_seli; an = s_aremn; }
            else             { sel = (s_selfb < 256) ? s_selfb : 0; an = 0.0f; }
            s_prefix = (s_prefix << 8) | (u32)sel;
            s_arem   = an;
        }
        __syncthreads();
    }
    if (tid == 0) {
        u32 tp = s_prefix;                  // smallest kept key under top-p
        s_tcut = (tp > tk) ? tp : tk;       // combined keep threshold
    }
    __syncthreads();
    const u32 tcut = s_tcut;

    // =========== Final pass: argmax over kept tokens of  x - log(e_i)  ===========
    float best = -__builtin_inff();
    int   bidx = 0x7FFFFFFF;
    for (int j = tid; j < VCOLS / 4; j += NTHREADS) {
        float4 v = p4[j];
        float xs[4] = {v.x, v.y, v.z, v.w};
#pragma unroll
        for (int c = 0; c < 4; ++c) {
            float x = xs[c] * invT;
            u32 key = monokey(x);
            if (key >= tcut) {
                int  i  = 4 * j + c;
                u32  gi = (u32)row * (u32)VCOLS + (u32)i;   // < 2^32
                float score = x - __logf(expo_of(gi));
                if (score > best || (score == best && i < bidx)) { best = score; bidx = i; }
            }
        }
    }
    // parallel argmax reduction with smallest-index tie-break
    s_redf[tid] = best;
    s_redi[tid] = bidx;
    __syncthreads();
    for (int off = NTHREADS / 2; off > 0; off >>= 1) {
        if (tid < off) {
            float ov = s_redf[tid + off]; int oi = s_redi[tid + off];
            if (ov > s_redf[tid] || (ov == s_redf[tid] && oi < s_redi[tid])) {
                s_redf[tid] = ov; s_redi[tid] = oi;
            }
        }
        __syncthreads();
    }
    if (tid == 0) out[row] = (float)s_redi[0];
}

extern "C" void kernel_launch(void* const* d_in, const int* in_sizes, int n_in,
                              void* d_out, int out_size, void* d_ws, size_t ws_size,
                              hipStream_t stream) {
    const float* logits = (const float*)d_in[0];
    const float* temps  = (const float*)d_in[1];
    const int*   tks    = (const int*)d_in[2];
    const float* tps    = (const float*)d_in[3];
    float* out = (float*)d_out;
    hipLaunchKernelGGL(sampler_kernel, dim3(BROWS), dim3(NTHREADS), 0, stream,
                       logits, temps, tks, tps, out);
}